// GCN_22299470201219
// MI455X (gfx1250) — compile-verified
//
#include <hip/hip_runtime.h>
#include <hip/hip_bf16.h>

typedef __attribute__((ext_vector_type(2))) float v2f;
typedef __attribute__((ext_vector_type(8))) float v8f;

// ---------------------------------------------------------------------------
// C[M,N] = op(A[M,K]) @ B[K,N], fp32, via V_WMMA_F32_16X16X4_F32.
// One wave per 16x16 output tile; 8 waves (256 thr) per block cover 8 N-tiles.
// RELU applies max(a,0) to the A operand on load (fuses relu(H) into GEMM2).
// ---------------------------------------------------------------------------
template <bool RELU>
__global__ __launch_bounds__(256)
void gcn_gemm_wmma_f32(const float* __restrict__ A, const float* __restrict__ B,
                       float* __restrict__ C, int M, int K, int N) {
  const int wave  = threadIdx.x >> 5;
  const int lane  = threadIdx.x & 31;
  const int laneN = lane & 15;            // M index (A) / N index (B,C)
  const int hi    = lane >> 4;            // 0: lanes 0-15, 1: lanes 16-31
  const int koff  = hi << 1;              // K sub-offset 0 or 2

  const int m0 = blockIdx.x * 16;
  const int n0 = (blockIdx.y * 8 + wave) * 16;
  if (n0 >= N) return;

  // Clamp A row so ragged M tiles read valid memory (garbage rows not stored).
  const int arow = min(m0 + laneN, M - 1);
  const float* __restrict__ Ap = A + (size_t)arow * K + koff;       // + k
  const float* __restrict__ Bp = B + (size_t)koff * N + n0 + laneN; // + k*N

  v8f acc = {};
#pragma unroll 8
  for (int k = 0; k < K; k += 4) {
    v2f a;
    a.x = Ap[k];
    a.y = Ap[k + 1];
    if (RELU) {
      a.x = fmaxf(a.x, 0.0f);
      a.y = fmaxf(a.y, 0.0f);
    }
    v2f b;
    b.x = Bp[(size_t)k * N];
    b.y = Bp[(size_t)(k + 1) * N];
    // D = A(16x4) * B(4x16) + C   (fp32, exact-precision accumulate)
    acc = __builtin_amdgcn_wmma_f32_16x16x4_f32(
        /*neg_a=*/false, a, /*neg_b=*/false, b,
        /*c_mod=*/(short)0, acc, /*reuse_a=*/false, /*reuse_b=*/false);
  }

  // C/D layout: VGPR j -> row (j + 8*hi), col laneN.
  const int crow = m0 + (hi << 3);
  float* __restrict__ Cp = C + (size_t)crow * N + n0 + laneN;
#pragma unroll
  for (int j = 0; j < 8; ++j) {
    if (crow + j < M) Cp[(size_t)j * N] = acc[j];
  }
}

// Four guaranteed-hardware non-returning fp32 adds (RMW at L2, STOREcnt path),
// sharing one 64-bit address VGPR pair with immediate byte offsets.
__device__ __forceinline__ void atomic_add_f32x4_noret(float* p, float a,
                                                       float b, float c,
                                                       float d) {
  asm volatile(
      "global_atomic_add_f32 %0, %1, off\n\t"
      "global_atomic_add_f32 %0, %2, off offset:4\n\t"
      "global_atomic_add_f32 %0, %3, off offset:8\n\t"
      "global_atomic_add_f32 %0, %4, off offset:12"
      :
      : "v"(p), "v"(a), "v"(b), "v"(c), "v"(d)
      : "memory");
}

// ---------------------------------------------------------------------------
// Y[row[e], :] += w[e] * X[col[e], :]   (segment-sum scatter, fp32 atomics).
// F/4 lanes per edge, float4 per lane -> fully coalesced gather of one row.
// ---------------------------------------------------------------------------
template <int F>
__global__ __launch_bounds__(256)
void gcn_spmm_scatter(const int* __restrict__ row, const int* __restrict__ col,
                      const float* __restrict__ w, const float* __restrict__ X,
                      float* __restrict__ Y, int nEdges) {
  constexpr int LPE = F / 4;  // lanes per edge (64 for F=256, 32 for F=128)
  const int e = blockIdx.x * (256 / LPE) + threadIdx.y;
  if (e >= nEdges) return;

  const int   r  = row[e];
  const int   c  = col[e];
  const float wt = w[e];
  const int   f  = threadIdx.x << 2;

  const float4 xv = *reinterpret_cast<const float4*>(X + (size_t)c * F + f);
  float* __restrict__ yp = Y + (size_t)r * F + f;
  atomic_add_f32x4_noret(yp, wt * xv.x, wt * xv.y, wt * xv.z, wt * xv.w);
}

// Y[i, f] = bias[f]  (bias folded into the scatter target's init; F power of 2)
__global__ __launch_bounds__(256)
void gcn_init_bias(float* __restrict__ Y, const float* __restrict__ bias,
                   int total, int Fmask) {
  const int i = blockIdx.x * 256 + threadIdx.x;
  if (i < total) Y[i] = bias[i & Fmask];
}

extern "C" void kernel_launch(void* const* d_in, const int* in_sizes, int n_in,
                              void* d_out, int out_size, void* d_ws, size_t ws_size,
                              hipStream_t stream) {
  constexpr int NFEAT = 256, NHID = 256, NOUT = 128;

  const float* feat = (const float*)d_in[0];
  const int*   row  = (const int*)d_in[1];
  const int*   col  = (const int*)d_in[2];
  const float* ew   = (const float*)d_in[3];
  const float* W1   = (const float*)d_in[4];
  const float* b1   = (const float*)d_in[5];
  const float* W2   = (const float*)d_in[6];
  const float* b2   = (const float*)d_in[7];
  float* out = (float*)d_out;

  const int nNodes = in_sizes[0] / NFEAT;
  const int nEdges = in_sizes[1];

  // Workspace: X1 [nNodes,256] | H [nNodes,256] ; X2 [nNodes,128] reuses X1.
  float* X1 = (float*)d_ws;
  float* H  = X1 + (size_t)nNodes * NHID;
  float* X2 = X1;

  const int mTiles = (nNodes + 15) / 16;

  // --- Layer 1 ---------------------------------------------------------
  // X1 = feat @ W1
  {
    dim3 grid(mTiles, NHID / 128);  // 8 N-tiles (waves) per block
    gcn_gemm_wmma_f32<false><<<grid, 256, 0, stream>>>(feat, W1, X1,
                                                       nNodes, NFEAT, NHID);
  }
  // H = b1 (broadcast), then H += scatter(ew * X1[col])
  {
    const int total = nNodes * NHID;
    gcn_init_bias<<<(total + 255) / 256, 256, 0, stream>>>(H, b1, total, NHID - 1);
  }
  {
    dim3 block(NHID / 4, 256 / (NHID / 4));  // (64,4): 4 edges per block
    const int epb = 256 / (NHID / 4);
    gcn_spmm_scatter<NHID><<<(nEdges + epb - 1) / epb, block, 0, stream>>>(
        row, col, ew, X1, H, nEdges);
  }

  // --- Layer 2 ---------------------------------------------------------
  // X2 = relu(H) @ W2   (ReLU fused into A-operand load)
  {
    dim3 grid(mTiles, (NOUT + 127) / 128);
    gcn_gemm_wmma_f32<true><<<grid, 256, 0, stream>>>(H, W2, X2,
                                                      nNodes, NHID, NOUT);
  }
  // out = b2 (broadcast), then out += scatter(ew * X2[col])
  {
    const int total = nNodes * NOUT;
    gcn_init_bias<<<(total + 255) / 256, 256, 0, stream>>>(out, b2, total, NOUT - 1);
  }
  {
    dim3 block(NOUT / 4, 256 / (NOUT / 4));  // (32,8): 8 edges per block
    const int epb = 256 / (NOUT / 4);
    gcn_spmm_scatter<NOUT><<<(nEdges + epb - 1) / epb, block, 0, stream>>>(
        row, col, ew, X2, out, nEdges);
  }
}